// SimpleGNN_12017318494531
// MI455X (gfx1250) — compile-verified
//
#include <hip/hip_runtime.h>

#define IN_DIM  128
#define HID_DIM 128
#define OUT_DIM 256
#define CAP     1024   // max slots: node 0 itself + its in-edges (Poisson mean ~16)

typedef float v2f __attribute__((ext_vector_type(2)));
typedef float v8f __attribute__((ext_vector_type(8)));

// ---------------------------------------------------------------- K0: zero ws
__global__ void k_zero(int* __restrict__ p, int n) {
    int i = blockIdx.x * blockDim.x + threadIdx.x;
    if (i < n) p[i] = 0;
}

// ------------------------------- K1: in-degrees + collect edges into node 0
__global__ void k_deg(const int* __restrict__ src, const int* __restrict__ dst, int E,
                      int* __restrict__ deg, int* __restrict__ cnt, int* __restrict__ tnode) {
    int e = blockIdx.x * blockDim.x + threadIdx.x;
    if (e >= E) return;
    int d = dst[e];
    atomicAdd(&deg[d], 1);
    if (d == 0) {
        int slot = atomicAdd(cnt, 1) + 1;          // slot 0 reserved for node 0 (self)
        if (slot < CAP) tnode[slot] = src[e];
    }
}

// ------------------------------------- K2: self-loop term per slot: dinv_v^2 * x[v]
__global__ void k_self(const float* __restrict__ x, const int* __restrict__ deg,
                       const int* __restrict__ cnt, const int* __restrict__ tnode,
                       float* __restrict__ agg1) {
    int nslots = min(*cnt + 1, CAP);
    int t = blockIdx.x;
    if (t >= nslots) return;
    int v = tnode[t];
    float dinv = rsqrtf((float)(deg[v] + 1));
    agg1[t * IN_DIM + threadIdx.x] += (dinv * dinv) * x[v * IN_DIM + threadIdx.x];
}

// --------------- K3: scan all edges; aggregate normed x[src] into matching slots
__global__ void k_gather1(const int* __restrict__ src, const int* __restrict__ dst, int E,
                          const int* __restrict__ deg, const int* __restrict__ cnt,
                          const int* __restrict__ tnode,
                          const float* __restrict__ x, float* __restrict__ agg1) {
    __shared__ int s_node[CAP];
    __shared__ int s_n;
    if (threadIdx.x == 0) s_n = min(*cnt + 1, CAP);
    __syncthreads();
    int n = s_n;
    for (int i = threadIdx.x; i < n; i += blockDim.x) s_node[i] = tnode[i];
    __syncthreads();

    int e = blockIdx.x * blockDim.x + threadIdx.x;
    if (e >= E) return;
    int d = dst[e];
    for (int t = 0; t < n; ++t) {
        if (s_node[t] == d) {                      // rare (~290 of 1.6M edges)
            int s = src[e];
            float dv = rsqrtf((float)(deg[d] + 1));
            float ds = rsqrtf((float)(deg[s] + 1));
            float w  = dv * ds;
            const float* xr = x + s * IN_DIM;
            float* ar = agg1 + t * IN_DIM;
            for (int k = 0; k < IN_DIM; ++k) atomicAdd(&ar[k], w * xr[k]);
        }
    }
}

// ------------- K4: h1 = relu(agg1 @ W1 + b1) via V_WMMA_F32_16X16X4_F32 tiles
__global__ __launch_bounds__(32)
void k_gemm1_wmma(const float* __restrict__ agg1, const float* __restrict__ W1,
                  const float* __restrict__ b1, const int* __restrict__ cnt,
                  float* __restrict__ h1) {
    int nslots = min(*cnt + 1, CAP);
    int r0 = blockIdx.x * 16;                      // slot-row tile
    if (r0 >= nslots) return;                      // wave-uniform: EXEC stays all-ones
    int n0 = blockIdx.y * 16;                      // output-col tile
    int lane = threadIdx.x;                        // 0..31, one wave32
    int half = lane >> 4;                          // 0: K pair {0,1}; 1: K pair {2,3}
    int lm   = lane & 15;

    v8f c = {};
    for (int k0 = 0; k0 < IN_DIM; k0 += 4) {
        int ka = k0 + half * 2;
        v2f a, b;
        // A 16x4 f32: lane half selects K sub-pair, lm = M
        const float* arow = agg1 + (r0 + lm) * IN_DIM + ka;
        a.x = arow[0];
        a.y = arow[1];
        // B 4x16 f32: lane half selects K sub-pair, lm = N
        const float* brow = W1 + ka * HID_DIM + n0 + lm;
        b.x = brow[0];
        b.y = brow[HID_DIM];
        c = __builtin_amdgcn_wmma_f32_16x16x4_f32(false, a, false, b,
                                                  (short)0, c, false, false);
    }
    // C/D layout: VGPR r -> M = r (+8 for upper half-lanes), N = lm
    float bias = b1[n0 + lm];
    for (int r = 0; r < 8; ++r) {
        int m = half ? (r + 8) : r;
        float v = c[r] + bias;
        h1[(r0 + m) * HID_DIM + n0 + lm] = v > 0.f ? v : 0.f;
    }
}

// --------- K5: agg2 = sum_t norm_t * h1[t]; out = agg2 @ W2 + b2  (one block)
__global__ void k_layer2(const float* __restrict__ h1, const float* __restrict__ W2,
                         const float* __restrict__ b2, const int* __restrict__ deg,
                         const int* __restrict__ cnt, const int* __restrict__ tnode,
                         float* __restrict__ out) {
    __shared__ float s_agg[HID_DIM];
    __shared__ float s_norm[CAP];
    int nslots = min(*cnt + 1, CAP);
    float dinv0 = rsqrtf((float)(deg[0] + 1));
    for (int t = threadIdx.x; t < nslots; t += blockDim.x) {
        int v = tnode[t];
        float dv = rsqrtf((float)(deg[v] + 1));
        s_norm[t] = dv * dinv0;                    // slot 0: v==0 -> dinv0^2
    }
    __syncthreads();
    if (threadIdx.x < HID_DIM) {
        float acc = 0.f;
        for (int t = 0; t < nslots; ++t)
            acc += s_norm[t] * h1[t * HID_DIM + threadIdx.x];
        s_agg[threadIdx.x] = acc;
    }
    __syncthreads();
    for (int j = threadIdx.x; j < OUT_DIM; j += blockDim.x) {
        float acc = b2[j];
        for (int k = 0; k < HID_DIM; ++k)
            acc += s_agg[k] * W2[k * OUT_DIM + j];
        out[j] = acc;
    }
}

extern "C" void kernel_launch(void* const* d_in, const int* in_sizes, int n_in,
                              void* d_out, int out_size, void* d_ws, size_t ws_size,
                              hipStream_t stream) {
    const float* x  = (const float*)d_in[0];
    const int*   ei = (const int*)d_in[1];
    const float* W1 = (const float*)d_in[2];
    const float* b1 = (const float*)d_in[3];
    const float* W2 = (const float*)d_in[4];
    const float* b2 = (const float*)d_in[5];
    int N = in_sizes[0] / IN_DIM;
    int E = in_sizes[1] / 2;
    const int* src = ei;
    const int* dst = ei + E;

    // workspace layout (contiguous 4-byte words so one zero-kernel covers init)
    int*   deg   = (int*)d_ws;                       // [N]
    int*   cnt   = deg + N;                          // [1]
    int*   tnode = cnt + 1;                          // [CAP] (zeroed -> tnode[0]=node 0)
    float* agg1  = (float*)(tnode + CAP);            // [CAP*IN_DIM]
    float* h1    = agg1 + (size_t)CAP * IN_DIM;      // [CAP*HID_DIM]

    int zero_words = N + 1 + CAP + CAP * IN_DIM;
    k_zero<<<(zero_words + 255) / 256, 256, 0, stream>>>((int*)d_ws, zero_words);
    k_deg<<<(E + 255) / 256, 256, 0, stream>>>(src, dst, E, deg, cnt, tnode);
    k_self<<<CAP, IN_DIM, 0, stream>>>(x, deg, cnt, tnode, agg1);
    k_gather1<<<(E + 255) / 256, 256, 0, stream>>>(src, dst, E, deg, cnt, tnode, x, agg1);
    dim3 g4(CAP / 16, HID_DIM / 16);
    k_gemm1_wmma<<<g4, 32, 0, stream>>>(agg1, W1, b1, cnt, h1);
    k_layer2<<<1, 256, 0, stream>>>(h1, W2, b2, deg, cnt, tnode, (float*)d_out);
}